// TextureMartingaleModule_46411416601254
// MI455X (gfx1250) — compile-verified
//
#include <hip/hip_runtime.h>
#include <hip/hip_bf16.h>
#include <math.h>

#define AS1 __attribute__((address_space(1)))
#define AS3 __attribute__((address_space(3)))

#if __has_builtin(__builtin_amdgcn_global_load_async_to_lds_b32)
#define HAVE_ASYNC_LDS 1
#else
#define HAVE_ASYNC_LDS 0
#endif

#define H_IMG 384
#define W_IMG 384
#define TILE_W 64
#define TILE_H 16
#define PAD_H (TILE_H + 2)   // 18
#define PAD_W (TILE_W + 2)   // 66
#define LDS_PITCH 68         // row pitch (floats), 16B-aligned rows
#define NTHREADS 256

// martingale with THETA=1: clip(exp(log(max(f,1e-5)) - 0.5), 1e-4, 1e4)
//                        = clip(max(f,1e-5) * e^-0.5, 1e-4, 1e4)
__device__ __forceinline__ float martingale(float f) {
    float m = fmaxf(f, 1e-5f) * 0.6065306597126334f;  // e^{-0.5}
    m = fminf(fmaxf(m, 1e-4f), 1e4f);
    return m;
}

__global__ __launch_bounds__(NTHREADS)
void TextureMartingaleModule_46411416601254_kernel(const float* __restrict__ x,
                                                   float* __restrict__ out) {
    __shared__ float xs[PAD_H * LDS_PITCH];  // raw taps (zero padded)
    __shared__ float es[PAD_H * LDS_PITCH];  // c*log(c) taps

    const int plane = blockIdx.z;                 // b*C + c  (0..127)
    const int x0 = blockIdx.x * TILE_W;
    const int y0 = blockIdx.y * TILE_H;
    const int tid = threadIdx.y * blockDim.x + threadIdx.x;

    const float* gin = x + (size_t)plane * (H_IMG * W_IMG);

    // ---- Stage padded tile into LDS.  OOB halo -> ds_store 0 (disjoint
    // addresses from the async writes, so no cross-counter ordering hazard);
    // in-bounds -> CDNA5 async global->LDS copy (ASYNCcnt).
    for (int idx = tid; idx < PAD_H * PAD_W; idx += NTHREADS) {
        const int py = idx / PAD_W;
        const int px = idx - py * PAD_W;
        const int gy = y0 + py - 1;
        const int gx = x0 + px - 1;
        const bool inb = (gy >= 0) & (gy < H_IMG) & (gx >= 0) & (gx < W_IMG);
        const int laddr = py * LDS_PITCH + px;
#if HAVE_ASYNC_LDS
        if (inb) {
            __builtin_amdgcn_global_load_async_to_lds_b32(
                (AS1 int*)(gin + (size_t)gy * W_IMG + gx),
                (AS3 int*)&xs[laddr], 0, 0);
        } else {
            xs[laddr] = 0.0f;
        }
#else
        xs[laddr] = inb ? gin[(size_t)gy * W_IMG + gx] : 0.0f;
#endif
    }
#if HAVE_ASYNC_LDS && __has_builtin(__builtin_amdgcn_s_wait_asynccnt)
    __builtin_amdgcn_s_wait_asynccnt(0);
#endif
    __syncthreads();

    // ---- Pointwise entropy transform once per staged element (1.16 logs/px
    // instead of 9 logs/px): e = max(v,1e-6) * ln(max(v,1e-6)).
    for (int idx = tid; idx < PAD_H * PAD_W; idx += NTHREADS) {
        const int py = idx / PAD_W;
        const int px = idx - py * PAD_W;
        const float v = xs[py * LDS_PITCH + px];
        const float c = fmaxf(v, 1e-6f);
        es[py * LDS_PITCH + px] = c * __logf(c);
    }
    __syncthreads();

    // ---- Per-pixel statistics.  blockDim = (64,4); thread owns 4 CONSECUTIVE
    // rows (4*ty .. 4*ty+3) and slides a 3-row register window vertically:
    // advancing a row costs 3 x-reads + 3 e-reads + row partials.
    const int tx = threadIdx.x;  // 0..63 -> tile column
    const int ty = threadIdx.y;  // 0..3
    const size_t HW = (size_t)H_IMG * W_IMG;
    float* outp = out + (size_t)plane * 4 * HW;

    const int py0 = 4 * ty;      // first output row; window rows py0..py0+2

    float xr[3][3];              // rolling 3x3 tap window
    float rs1[3], rs2[3], er[3]; // rolling per-row partials: sum x, sum x^2, sum e

#pragma unroll
    for (int r = 0; r < 3; ++r) {
        const int rowb = (py0 + r) * LDS_PITCH + tx;
        const float a = xs[rowb + 0], b = xs[rowb + 1], c = xs[rowb + 2];
        xr[r][0] = a; xr[r][1] = b; xr[r][2] = c;
        rs1[r] = a + b + c;
        rs2[r] = fmaf(a, a, fmaf(b, b, c * c));
        er[r]  = es[rowb + 0] + es[rowb + 1] + es[rowb + 2];
    }

#pragma unroll
    for (int i = 0; i < 4; ++i) {
        const float S1 = rs1[0] + rs1[1] + rs1[2];
        const float S2 = rs2[0] + rs2[1] + rs2[2];
        const float SE = er[0] + er[1] + er[2];
        const float mean = S1 * (1.0f / 9.0f);

        // sum (x-mean)^2 = S2 - S1*mean  (exact algebra; clamp fp residue)
        const float Svar = fmaxf(fmaf(-S1, mean, S2), 0.0f);

        float Sabs = 0.f;
#pragma unroll
        for (int r = 0; r < 3; ++r)
#pragma unroll
            for (int c = 0; c < 3; ++c)
                Sabs += fabsf(xr[r][c] - mean);

        float stdv = sqrtf(Svar * (1.0f / 8.0f));
        stdv = fmaxf(stdv, 0.001f);
        const float contrast = (Svar * (1.0f / 9.0f)) / (stdv * stdv);
        const float energy   = S2 * (1.0f / 9.0f);
        const float entropy  = -SE * (1.0f / 9.0f);
        const float homog    = 1.0f / (1.0f + Sabs * (1.0f / 9.0f) + 1e-6f);

        const size_t oidx = (size_t)(y0 + py0 + i) * W_IMG + (x0 + tx);
        outp[0 * HW + oidx] = martingale(contrast);
        outp[1 * HW + oidx] = martingale(energy);
        outp[2 * HW + oidx] = martingale(entropy);
        outp[3 * HW + oidx] = martingale(homog);

        if (i < 3) {
            // slide window down one row
#pragma unroll
            for (int r = 0; r < 2; ++r) {
                xr[r][0] = xr[r + 1][0]; xr[r][1] = xr[r + 1][1]; xr[r][2] = xr[r + 1][2];
                rs1[r] = rs1[r + 1]; rs2[r] = rs2[r + 1]; er[r] = er[r + 1];
            }
            const int rowb = (py0 + 3 + i) * LDS_PITCH + tx;
            const float a = xs[rowb + 0], b = xs[rowb + 1], c = xs[rowb + 2];
            xr[2][0] = a; xr[2][1] = b; xr[2][2] = c;
            rs1[2] = a + b + c;
            rs2[2] = fmaf(a, a, fmaf(b, b, c * c));
            er[2]  = es[rowb + 0] + es[rowb + 1] + es[rowb + 2];
        }
    }
}

extern "C" void kernel_launch(void* const* d_in, const int* in_sizes, int n_in,
                              void* d_out, int out_size, void* d_ws, size_t ws_size,
                              hipStream_t stream) {
    (void)n_in; (void)out_size; (void)d_ws; (void)ws_size;
    const float* x = (const float*)d_in[0];
    float* out = (float*)d_out;

    const int planes = in_sizes[0] / (H_IMG * W_IMG);  // b*c = 128
    dim3 block(TILE_W, NTHREADS / TILE_W, 1);          // (64,4) = 8 wave32s
    dim3 grid(W_IMG / TILE_W, H_IMG / TILE_H, planes); // (6,24,128)
    TextureMartingaleModule_46411416601254_kernel<<<grid, block, 0, stream>>>(x, out);
}